// Dynamic_Adaptive_Scan_19834158973075
// MI455X (gfx1250) — compile-verified
//
#include <hip/hip_runtime.h>
#include <math.h>

#define CCH 512
#define GRP 16
#define GCH 32
#define HH 64
#define WW 64
#define EPSV 1e-6f

typedef __attribute__((ext_vector_type(2))) float v2f;
typedef __attribute__((ext_vector_type(8))) float v8f;

__global__ __launch_bounds__(256) void das_fused(
    const float* __restrict__ in_first,   // [N,512,64,64]
    const float* __restrict__ in_last,    // [N,64,64,512]
    const float* __restrict__ dw_w,       // [512,1,3,3]
    const float* __restrict__ dw_b,       // [512]
    const float* __restrict__ ln_g,       // [512]
    const float* __restrict__ ln_b,       // [512]
    const float* __restrict__ off_w,      // [32,512]
    const float* __restrict__ off_b,      // [32]
    float* __restrict__ out)              // [N,512,64,64]
{
    __shared__ float xbuf[WW][CCH + 1];     // 131,328 B, padded -> conflict-free
    __shared__ float offbuf[WW][2 * GRP];   // 8,192 B

    const int t    = threadIdx.x;
    const int bid  = blockIdx.x;
    const int n    = bid / HH;
    const int h    = bid % HH;
    const int lane = t & 31;
    const int wv   = t >> 5;                // wave id 0..7

    // ---- Phase 1: depthwise 3x3 conv + bias for row h, all 512 channels ----
    {
        const int w    = t & 63;
        const int csub = t >> 6;            // 0..3
        for (int cb = 0; cb < CCH / 4; ++cb) {
            const int c = cb * 4 + csub;
            const float* wp  = dw_w + c * 9;
            const float* src = in_first + (size_t)(n * CCH + c) * (HH * WW);
            float acc = dw_b[c];
            #pragma unroll
            for (int ky = 0; ky < 3; ++ky) {
                const int y = h + ky - 1;
                if (y < 0 || y >= HH) continue;
                #pragma unroll
                for (int kx = 0; kx < 3; ++kx) {
                    const int x = w + kx - 1;
                    if (x < 0 || x >= WW) continue;
                    acc = fmaf(src[y * WW + x], wp[ky * 3 + kx], acc);
                }
            }
            xbuf[w][c] = acc;
        }
    }
    __syncthreads();

    // ---- Phase 2: LayerNorm over channels + exact GELU (wave per 8 pixels) ----
    {
        for (int pw = 0; pw < 8; ++pw) {
            const int w = wv * 8 + pw;
            float s = 0.f, s2 = 0.f;
            float v[16];
            #pragma unroll
            for (int j = 0; j < 16; ++j) {
                const float x = xbuf[w][lane + 32 * j];
                v[j] = x;
                s += x;
                s2 = fmaf(x, x, s2);
            }
            #pragma unroll
            for (int m = 16; m >= 1; m >>= 1) {
                s  += __shfl_xor(s,  m);
                s2 += __shfl_xor(s2, m);
            }
            const float mu   = s * (1.f / CCH);
            const float var  = s2 * (1.f / CCH) - mu * mu;
            const float rstd = rsqrtf(var + EPSV);
            #pragma unroll
            for (int j = 0; j < 16; ++j) {
                const int c = lane + 32 * j;
                float xh = (v[j] - mu) * rstd * ln_g[c] + ln_b[c];
                xh = 0.5f * xh * (1.f + erff(xh * 0.70710678118654752f)); // exact GELU
                xbuf[w][c] = xh;
            }
        }
    }
    __syncthreads();

    // ---- Phase 3: offset GEMM [64w x 512c] x [512c x 32o] via V_WMMA_F32_16X16X4_F32 ----
    {
        const int wtile = wv >> 1;            // 0..3 : block of 16 pixels (M)
        const int otile = wv & 1;             // 0..1 : block of 16 outputs (N)
        const int half  = lane >> 4;          // selects K {0,1} vs {2,3}
        const int lr    = lane & 15;
        const int wrow  = wtile * 16 + lr;    // A-fragment row (M)
        const int ocol  = otile * 16 + lr;    // B-fragment col (N)
        const float* bw = off_w + (size_t)ocol * CCH;

        v8f acc = {0.f, 0.f, 0.f, 0.f, 0.f, 0.f, 0.f, 0.f};
        for (int k0 = 0; k0 < CCH; k0 += 4) {
            v2f a, b;
            a.x = xbuf[wrow][k0 + 2 * half];
            a.y = xbuf[wrow][k0 + 2 * half + 1];
            b.x = bw[k0 + 2 * half];
            b.y = bw[k0 + 2 * half + 1];
            acc = __builtin_amdgcn_wmma_f32_16x16x4_f32(
                false, a, false, b, (short)0, acc, false, false);
        }
        const float bias = off_b[ocol];
        #pragma unroll
        for (int vv = 0; vv < 8; ++vv) {
            const int w = wtile * 16 + vv + 8 * half;  // D row per C/D VGPR layout
            offbuf[w][ocol] = acc[vv] + bias;
        }
    }
    __syncthreads();

    // ---- Phase 4: DCNv3 bilinear sampling (lane = channel-in-group, coalesced taps) ----
    {
        const int cc = lane;
        const float* src = in_last + (size_t)n * (HH * WW * CCH);
        for (int i = 0; i < 128; ++i) {
            const int p = wv * 128 + i;
            const int w = p >> 4;
            const int g = p & 15;
            const float px = (float)w + offbuf[w][2 * g];       // OFFSET_SCALE = 1
            const float py = (float)h + offbuf[w][2 * g + 1];
            const float x0  = floorf(px);
            const float y0  = floorf(py);
            const float wx1 = px - x0;
            const float wy1 = py - y0;
            float sum = 0.f;
            #pragma unroll
            for (int ty = 0; ty < 2; ++ty) {
                #pragma unroll
                for (int tx = 0; tx < 2; ++tx) {
                    const float xf = x0 + (float)tx;
                    const float yf = y0 + (float)ty;
                    const float wgt = (tx ? wx1 : 1.f - wx1) * (ty ? wy1 : 1.f - wy1);
                    if (xf >= 0.f && xf <= (float)(WW - 1) &&
                        yf >= 0.f && yf <= (float)(HH - 1)) {
                        const int xi = (int)xf;
                        const int yi = (int)yf;
                        sum = fmaf(src[((size_t)yi * WW + xi) * CCH + g * GCH + cc],
                                   wgt, sum);
                    }
                }
            }
            xbuf[w][g * GCH + cc] = sum;   // reuse row buffer
        }
    }
    __syncthreads();

    // ---- Phase 5: coalesced NCHW writeout ----
    {
        const int w    = t & 63;
        const int csub = t >> 6;
        float* dst = out + (size_t)n * CCH * (HH * WW) + (size_t)h * WW;
        for (int cb = 0; cb < CCH / 4; ++cb) {
            const int c = cb * 4 + csub;
            dst[(size_t)c * (HH * WW) + w] = xbuf[w][c];
        }
    }
}

extern "C" void kernel_launch(void* const* d_in, const int* in_sizes, int n_in,
                              void* d_out, int out_size, void* d_ws, size_t ws_size,
                              hipStream_t stream) {
    const float* in_first = (const float*)d_in[0];
    const float* in_last  = (const float*)d_in[1];
    const float* dw_w     = (const float*)d_in[2];
    const float* dw_b     = (const float*)d_in[3];
    const float* ln_g     = (const float*)d_in[4];
    const float* ln_b     = (const float*)d_in[5];
    const float* off_w    = (const float*)d_in[6];
    const float* off_b    = (const float*)d_in[7];
    float* out = (float*)d_out;

    const int N = 8;
    das_fused<<<dim3(N * HH), dim3(256), 0, stream>>>(
        in_first, in_last, dw_w, dw_b, ln_g, ln_b, off_w, off_b, out);
}